// PartsEmbeddingsEMA_25013889532442
// MI455X (gfx1250) — compile-verified
//
#include <hip/hip_runtime.h>
#include <stdint.h>

// Reverse-time EMA scan, B=16 N=128 T=64 P=6 E=128.
// One wave32 per (b,n) chain. Masked timesteps skipped entirely (~halves HBM
// traffic). embs t-blocks staged into LDS with CDNA5
// global_load_async_to_lds_b128, 4-deep pipelined per wave (12 KB in flight
// per wave so 2048 waves can saturate 23.3 TB/s), ASYNCcnt-synchronized.

#define EMA_ALPHA 0.9f

constexpr int T_DIM  = 64;
constexpr int P_DIM  = 6;
constexpr int E_DIM  = 128;
constexpr int PE     = P_DIM * E_DIM;       // 768 floats = 3072 bytes per t-block
constexpr int TBYTES = PE * 4;              // 3072
constexpr int CHAINS = 16 * 128;            // 2048
constexpr int WAVES_PER_BLOCK = 4;          // 4 chains per 128-thread block
constexpr int DEPTH  = 4;                   // async pipeline depth (buffers/wave)

// Pop highest set bit (uniform across the wave).
__device__ __forceinline__ int pop_msb(uint64_t& b) {
  if (b == 0ull) return -1;
  int t = 63 - __builtin_clzll(b);
  b &= ~(1ull << t);
  return t;
}

// Stage one 3072-byte t-block into LDS: 6x async B128, 512 B per instruction.
// ISA: the immediate offset is added to both the LDS and the global address.
__device__ __forceinline__ void stage_async(uint32_t lds_addr, const char* gaddr) {
  asm volatile(
      "global_load_async_to_lds_b128 %0, %1, off offset:0\n\t"
      "global_load_async_to_lds_b128 %0, %1, off offset:512\n\t"
      "global_load_async_to_lds_b128 %0, %1, off offset:1024\n\t"
      "global_load_async_to_lds_b128 %0, %1, off offset:1536\n\t"
      "global_load_async_to_lds_b128 %0, %1, off offset:2048\n\t"
      "global_load_async_to_lds_b128 %0, %1, off offset:2560"
      :: "v"(lds_addr), "v"(gaddr) : "memory");
}

// Async loads complete in order: waiting <= 6*k leaves the youngest k groups
// outstanding, guaranteeing everything older is resident in LDS.
__device__ __forceinline__ void wait_async_pending(int younger) {
  if (younger >= 3)      asm volatile("s_wait_asynccnt 18" ::: "memory");
  else if (younger == 2) asm volatile("s_wait_asynccnt 12" ::: "memory");
  else if (younger == 1) asm volatile("s_wait_asynccnt 6"  ::: "memory");
  else                   asm volatile("s_wait_asynccnt 0"  ::: "memory");
}
__device__ __forceinline__ void wait_ds_0() {
  asm volatile("s_wait_dscnt 0" ::: "memory");  // LDS reads retired before reuse
}

__global__ __launch_bounds__(WAVES_PER_BLOCK * 32)
void PartsEmbeddingsEMA_kernel(const float* __restrict__ embs,
                               const float* __restrict__ vis,
                               const unsigned char* __restrict__ masks,
                               float* __restrict__ out) {
  __shared__ __align__(16) float lds_buf[WAVES_PER_BLOCK * DEPTH * PE];  // 48 KB

  const int wv    = (int)(threadIdx.x >> 5);
  const int lane  = (int)(threadIdx.x & 31);
  const int chain = (int)blockIdx.x * WAVES_PER_BLOCK + wv;
  if (chain >= CHAINS) return;  // never taken with exact grid; uniform anyway

  // Lane covers 16 bytes; lane's k-th float4 is chunk (lane + 32k) => part
  // index p == k for every lane (E=128 floats per part, 32 chunks per part).
  const char*  eb   = (const char*)embs + (size_t)chain * T_DIM * TBYTES + (size_t)lane * 16;
  const float* vrow = vis + (size_t)chain * T_DIM * P_DIM;

  // Compress the chain's 64 mask bytes (0/1) into one 64-bit bitmask.
  uint64_t mb = 0;
  {
    const uint32_t* m32 = (const uint32_t*)(masks + (size_t)chain * T_DIM);
#pragma unroll
    for (int i = 0; i < 16; ++i) {
      uint32_t w   = m32[i];
      uint32_t nib = (w | (w >> 7) | (w >> 14) | (w >> 21)) & 0xFu;
      mb |= (uint64_t)nib << (4 * i);
    }
  }

  float* mybase = &lds_buf[(size_t)wv * DEPTH * PE];
  const uint32_t lbase = (uint32_t)(uintptr_t)mybase + (uint32_t)lane * 16u;

  float4 carry[6];
  float  visc[6];
#pragma unroll
  for (int k = 0; k < 6; ++k) {
    carry[k] = make_float4(0.f, 0.f, 0.f, 0.f);
    visc[k]  = 0.f;
  }

  // 4-slot software pipeline: tA (oldest, next to consume) .. tD (youngest).
  float vA[6], vB[6], vC[6], vD[6];
#pragma unroll
  for (int k = 0; k < 6; ++k) { vA[k] = 0.f; vB[k] = 0.f; vC[k] = 0.f; vD[k] = 0.f; }

  int tA = pop_msb(mb);
  int tB = pop_msb(mb);
  int tC = pop_msb(mb);
  int tD = pop_msb(mb);
  if (tA >= 0) {
    stage_async(lbase + 0 * TBYTES, eb + (size_t)tA * TBYTES);
#pragma unroll
    for (int k = 0; k < 6; ++k) vA[k] = vrow[tA * P_DIM + k];
  }
  if (tB >= 0) {
    stage_async(lbase + 1 * TBYTES, eb + (size_t)tB * TBYTES);
#pragma unroll
    for (int k = 0; k < 6; ++k) vB[k] = vrow[tB * P_DIM + k];
  }
  if (tC >= 0) {
    stage_async(lbase + 2 * TBYTES, eb + (size_t)tC * TBYTES);
#pragma unroll
    for (int k = 0; k < 6; ++k) vC[k] = vrow[tC * P_DIM + k];
  }
  if (tD >= 0) {
    stage_async(lbase + 3 * TBYTES, eb + (size_t)tD * TBYTES);
#pragma unroll
    for (int k = 0; k < 6; ++k) vD[k] = vrow[tD * P_DIM + k];
  }

  int buf = 0;  // cyclic 0..3; slot tA lives in buffer `buf`
  while (tA >= 0) {
    const int younger = (tB >= 0) + (tC >= 0) + (tD >= 0);
    wait_async_pending(younger);

    // Per-part scalar weights (6 parts, redundant per lane — cheap).
    float trk[6], det[6];
#pragma unroll
    for (int k = 0; k < 6; ++k) {
      float vc = visc[k], vt = vA[k];
      float x  = ((vc != 0.f) != (vt != 0.f)) ? 1.f : 0.f;
      trk[k]   = vc * vt * EMA_ALPHA + x * vc;
      det[k]   = vc * vt * (1.f - EMA_ALPHA) + x * vt;
      visc[k]  = fmaxf(vc, vt);
    }

    // Consume the staged block from LDS: 6x ds_load_b128 + 24 FMAs per lane.
    const float4* src = (const float4*)(mybase + buf * PE);
#pragma unroll
    for (int k = 0; k < 6; ++k) {
      float4 e   = src[lane + 32 * k];
      carry[k].x = trk[k] * carry[k].x + det[k] * e.x;
      carry[k].y = trk[k] * carry[k].y + det[k] * e.y;
      carry[k].z = trk[k] * carry[k].z + det[k] * e.z;
      carry[k].w = trk[k] * carry[k].w + det[k] * e.w;
    }

    // Refill the buffer we just drained with the next valid timestep.
    float vE[6];
#pragma unroll
    for (int k = 0; k < 6; ++k) vE[k] = 0.f;
    int tE = pop_msb(mb);
    if (tE >= 0) {
      wait_ds_0();  // our ds reads of this buffer must retire before overwrite
      stage_async(lbase + (uint32_t)buf * TBYTES, eb + (size_t)tE * TBYTES);
#pragma unroll
      for (int k = 0; k < 6; ++k) vE[k] = vrow[tE * P_DIM + k];
    }

    tA = tB; tB = tC; tC = tD; tD = tE;
#pragma unroll
    for (int k = 0; k < 6; ++k) {
      vA[k] = vB[k]; vB[k] = vC[k]; vC[k] = vD[k]; vD[k] = vE[k];
    }
    buf = (buf + 1) & (DEPTH - 1);
  }

  // Final carry -> out[chain, 768], coalesced B128 stores.
  float4* o = (float4*)out + (size_t)chain * (PE / 4);
#pragma unroll
  for (int k = 0; k < 6; ++k) o[lane + 32 * k] = carry[k];
}

extern "C" void kernel_launch(void* const* d_in, const int* in_sizes, int n_in,
                              void* d_out, int out_size, void* d_ws, size_t ws_size,
                              hipStream_t stream) {
  (void)in_sizes; (void)n_in; (void)d_ws; (void)ws_size; (void)out_size;
  const float*         embs  = (const float*)d_in[0];
  const float*         vis   = (const float*)d_in[1];
  const unsigned char* masks = (const unsigned char*)d_in[2];  // jax bool = 1 byte
  float*               out   = (float*)d_out;

  dim3 grid(CHAINS / WAVES_PER_BLOCK);   // 512 blocks
  dim3 block(WAVES_PER_BLOCK * 32);      // 128 threads = 4 wave32
  hipLaunchKernelGGL(PartsEmbeddingsEMA_kernel, grid, block, 0, stream,
                     embs, vis, masks, out);
}